// RotaryEmbedding_63591285784838
// MI455X (gfx1250) — compile-verified
//
#include <hip/hip_runtime.h>
#include <stdint.h>

// RoPE: x:(4,4096,32,128) f32, cos/sin:(4096,64) f32 -> out same shape as x.
// Memory-bound streaming kernel:
//  - one block per (b,s) slice (4096 contiguous floats) -> s is block-uniform
//  - wave 0 stages the cos/sin row (64+64 floats) into LDS via
//    global_load_async_to_lds_b64 (CDNA5 async path, ASYNCcnt)
//  - x/out moved with 128-bit non-temporal loads/stores (streamed once,
//    512MB total > 192MB L2, so NT keeps L2 for the hot tables)

typedef float f4 __attribute__((ext_vector_type(4)));

__global__ __launch_bounds__(256) void rope_fwd_kernel(
    const float* __restrict__ x,
    const float* __restrict__ cosp,
    const float* __restrict__ sinp,
    float* __restrict__ out)
{
    // [0..63] = cos row, [64..127] = sin row
    __shared__ __align__(16) float tile[128];

    const unsigned blk = blockIdx.x;     // blk = b*4096 + s
    const unsigned t   = threadIdx.x;    // 0..255
    const unsigned s   = blk & 4095u;    // seq position (uniform per block)

    if (t < 32u) {
        // Wave 0: async-copy cos/sin row into LDS. 32 lanes x 8B = 256B per row.
        const float* crow = cosp + (size_t)s * 64u;
        const float* srow = sinp + (size_t)s * 64u;
        // Low 32 bits of a flat pointer to LDS == wave-relative LDS byte offset.
        unsigned lds_base = (unsigned)(size_t)(&tile[0]);
        unsigned ldsc = lds_base + t * 8u;
        unsigned ldss = lds_base + 256u + t * 8u;
        unsigned voff = t * 8u;
        asm volatile("global_load_async_to_lds_b64 %0, %1, %2"
                     :: "v"(ldsc), "v"(voff), "s"(crow) : "memory");
        asm volatile("global_load_async_to_lds_b64 %0, %1, %2"
                     :: "v"(ldss), "v"(voff), "s"(srow) : "memory");
        asm volatile("s_wait_asynccnt 0" ::: "memory");
    }
    __syncthreads();

    // Each thread handles 4 float4 chunks at stride 1024 floats; since
    // 1024 % 128 == 0, the feature offset d (and thus cos/sin) is identical
    // for all 4 chunks -> one LDS read pair serves 4 chunks.
    const unsigned d  = (t * 4u) & 127u;   // even, 0..124
    const unsigned i0 = d >> 1;            // pair index, even, 0..62
    const float c0 = tile[i0];
    const float c1 = tile[i0 + 1u];
    const float s0 = tile[64u + i0];
    const float s1 = tile[64u + i0 + 1u];

    size_t e = (size_t)blk * 4096u + (size_t)t * 4u;
#pragma unroll
    for (int k = 0; k < 4; ++k) {
        f4 v = __builtin_nontemporal_load((const f4*)(x + e));
        f4 o;
        o.x = v.x * c0 - v.y * s0;
        o.y = v.x * s0 + v.y * c0;
        o.z = v.z * c1 - v.w * s1;
        o.w = v.z * s1 + v.w * c1;
        __builtin_nontemporal_store(o, (f4*)(out + e));
        e += 1024u;
    }
}

extern "C" void kernel_launch(void* const* d_in, const int* in_sizes, int n_in,
                              void* d_out, int out_size, void* d_ws, size_t ws_size,
                              hipStream_t stream) {
    (void)n_in; (void)d_ws; (void)ws_size;
    const float* x    = (const float*)d_in[0];
    const float* cosp = (const float*)d_in[1];
    const float* sinp = (const float*)d_in[2];
    float* out = (float*)d_out;

    // x has B*S*H*D = out_size elements; one block per (b,s) slice of 4096.
    const int total  = out_size > 0 ? out_size : in_sizes[0];
    const int blocks = total / 4096;  // = 4 * 4096 = 16384 for the reference
    rope_fwd_kernel<<<blocks, 256, 0, stream>>>(x, cosp, sinp, out);
}